// SingleHeadNodeAttention_35605278884024
// MI455X (gfx1250) — compile-verified
//
#include <hip/hip_runtime.h>
#include <hip/hip_bf16.h>

#define N_NODES 100000
#define N_EDGES 1600000
#define D 64
#define ROW_TILES (N_NODES / 16)   // 6250 exactly

typedef float v2f __attribute__((ext_vector_type(2)));
typedef float v8f __attribute__((ext_vector_type(8)));

// ---------------------------------------------------------------------------
// Phase 0: re-initialize workspace + output every launch (deterministic).
// ---------------------------------------------------------------------------
__global__ void init_ws_kernel(unsigned int* __restrict__ segmax_bits,
                               float* __restrict__ denom,
                               float* __restrict__ out) {
    int t = blockIdx.x * blockDim.x + threadIdx.x;
    if (t < N_NODES) {
        // monotone map of -inf (0xFF800000): negative -> ~v
        segmax_bits[t] = 0x007FFFFFu;
        denom[t] = 0.0f;
    }
    if (t < N_NODES * D) out[t] = 0.0f;
}

// ---------------------------------------------------------------------------
// Phase 1: Q/K/V = X @ W^T via V_WMMA_F32_16X16X4_F32.
// out[n,o] = sum_k X[n,k] * W[o,k].  A = X tile (16x4, MxK), B = W^T tile
// (4x16, KxN) -> B[k][n] = W[n][k]; since both X and W are row-major with K
// contiguous, both fragments load as float2 at [row*64 + kb].
//
// fp32 A 16x4 layout (ISA 7.12.2): lanes 0-15 M=0..15 {K=kb,kb+1},
//                                  lanes 16-31 M=0..15 {K=kb+2,kb+3}.
// fp32 C/D 16x16 layout: VGPR r -> lanes 0-15 M=r, lanes 16-31 M=r+8; N=lane&15.
// One wave computes one 16x16 tile; block = 4 waves = 4 column tiles.
// grid = (6250 row tiles, 3 matrices).
// ---------------------------------------------------------------------------
__global__ void qkv_gemm_kernel(const float* __restrict__ X,
                                const float* __restrict__ Wq,
                                const float* __restrict__ Wk,
                                const float* __restrict__ Wv,
                                float* __restrict__ Q,
                                float* __restrict__ K,
                                float* __restrict__ V) {
    const int lane = threadIdx.x & 31;
    const int wave = threadIdx.x >> 5;          // column tile 0..3
    const int rowTile = blockIdx.x;             // 0..6249
    const int mat = blockIdx.y;                 // 0=Q 1=K 2=V

    const float* __restrict__ W = (mat == 0) ? Wq : (mat == 1) ? Wk : Wv;
    float* __restrict__ O       = (mat == 0) ? Q  : (mat == 1) ? K  : V;

    const int m  = lane & 15;                   // row/col within tile
    const int kh = lane >> 4;                   // K half-select (0/1)
    const long rowA = (long)(rowTile * 16 + m) * D;
    const long rowB = (long)(wave * 16 + m) * D;

    v8f acc = {};
#pragma unroll
    for (int kk = 0; kk < 16; ++kk) {
        const int kb = kk * 4 + kh * 2;
        v2f a = *(const v2f*)(X + rowA + kb);   // A[m][kb..kb+1]
        v2f b = *(const v2f*)(W + rowB + kb);   // (W^T)[kb..kb+1][col]
        acc = __builtin_amdgcn_wmma_f32_16x16x4_f32(
            /*neg_a=*/false, a, /*neg_b=*/false, b,
            /*c_mod=*/(short)0, acc, /*reuse_a=*/false, /*reuse_b=*/false);
    }

    const int colBase = wave * 16 + m;
#pragma unroll
    for (int r = 0; r < 8; ++r) {
        O[(long)(rowTile * 16 + r + 8 * kh) * D + colBase] = acc[r];
    }
}

// ---------------------------------------------------------------------------
// Phase 2: per-edge score = dot(Q[dst], K[src]) / 8, segment-max by dst.
// 8 lanes per edge, each covering 8 contiguous floats (two b128 loads).
// Float segment-max via monotone uint mapping + atomicMax.
// ---------------------------------------------------------------------------
__device__ __forceinline__ unsigned int float_to_ord(float f) {
    unsigned int v = __float_as_uint(f);
    return (v & 0x80000000u) ? ~v : (v | 0x80000000u);
}
__device__ __forceinline__ float ord_to_float(unsigned int u) {
    return (u & 0x80000000u) ? __uint_as_float(u & 0x7FFFFFFFu)
                             : __uint_as_float(~u);
}

__global__ void edge_score_kernel(const float* __restrict__ Q,
                                  const float* __restrict__ K,
                                  const int* __restrict__ src,
                                  const int* __restrict__ dst,
                                  float* __restrict__ score,
                                  unsigned int* __restrict__ segmax_bits) {
    const int t = blockIdx.x * blockDim.x + threadIdx.x;
    const int e = t >> 3;
    const int l = t & 7;
    if (e >= N_EDGES) return;

    const int s = src[e];
    const int d = dst[e];

    const float4* q = (const float4*)(Q + (long)d * D) + l * 2;
    const float4* k = (const float4*)(K + (long)s * D) + l * 2;
    float4 q0 = q[0], q1 = q[1];
    float4 k0 = k[0], k1 = k[1];

    float p = q0.x * k0.x + q0.y * k0.y + q0.z * k0.z + q0.w * k0.w
            + q1.x * k1.x + q1.y * k1.y + q1.z * k1.z + q1.w * k1.w;

    p += __shfl_xor(p, 4, 8);
    p += __shfl_xor(p, 2, 8);
    p += __shfl_xor(p, 1, 8);

    if (l == 0) {
        const float sc = p * 0.125f;            // 1/sqrt(64)
        score[e] = sc;
        atomicMax(segmax_bits + d, float_to_ord(sc));
    }
}

// ---------------------------------------------------------------------------
// Phase 3: fused softmax-weight + scatter.  Accumulate UNNORMALIZED
// w*V[src] into out[dst] and w into denom[dst]; normalization deferred to
// Phase 4 (avoids a third edge sweep).  16 lanes/edge, 4 floats each.
// ---------------------------------------------------------------------------
__global__ void edge_aggregate_kernel(const float* __restrict__ V,
                                      const int* __restrict__ src,
                                      const int* __restrict__ dst,
                                      const float* __restrict__ score,
                                      const unsigned int* __restrict__ segmax_bits,
                                      float* __restrict__ denom,
                                      float* __restrict__ out) {
    const int t = blockIdx.x * blockDim.x + threadIdx.x;
    const int e = t >> 4;
    const int l = t & 15;
    if (e >= N_EDGES) return;

    const int s = src[e];
    const int d = dst[e];

    const float m = ord_to_float(segmax_bits[d]);
    const float w = __expf(score[e] - m);

    if (l == 0) atomicAdd(denom + d, w);

    float4 v = *((const float4*)(V + (long)s * D) + l);
    float* o = out + (long)d * D + l * 4;
    atomicAdd(o + 0, w * v.x);
    atomicAdd(o + 1, w * v.y);
    atomicAdd(o + 2, w * v.z);
    atomicAdd(o + 3, w * v.w);
}

// ---------------------------------------------------------------------------
// Phase 4: normalize.  Nodes with no in-edges keep 0 (guard denom==0).
// ---------------------------------------------------------------------------
__global__ void finalize_kernel(float* __restrict__ out,
                                const float* __restrict__ denom) {
    const int t = blockIdx.x * blockDim.x + threadIdx.x;
    if (t >= N_NODES * D) return;
    const float dn = denom[t >> 6];
    out[t] *= (dn > 0.0f) ? (1.0f / dn) : 0.0f;
}

// ---------------------------------------------------------------------------
extern "C" void kernel_launch(void* const* d_in, const int* in_sizes, int n_in,
                              void* d_out, int out_size, void* d_ws, size_t ws_size,
                              hipStream_t stream) {
    const float* X   = (const float*)d_in[0];   // node_feats [N, 64]
    const int*   src = (const int*)d_in[1];     // [E]
    const int*   dst = (const int*)d_in[2];     // [E]
    const float* Wq  = (const float*)d_in[3];   // [64, 64]
    const float* Wk  = (const float*)d_in[4];
    const float* Wv  = (const float*)d_in[5];
    float* out = (float*)d_out;                 // [N, 64]

    // Workspace layout (all fp32-aligned): Q | K | V | score | segmax | denom
    float* Q = (float*)d_ws;
    float* K = Q + (size_t)N_NODES * D;
    float* V = K + (size_t)N_NODES * D;
    float* score = V + (size_t)N_NODES * D;
    unsigned int* segmax = (unsigned int*)(score + (size_t)N_EDGES);
    float* denom = (float*)(segmax + (size_t)N_NODES);

    // Phase 0: init (covers out zeroing + segmax/denom reset)
    {
        const int total = N_NODES * D;
        init_ws_kernel<<<(total + 255) / 256, 256, 0, stream>>>(segmax, denom, out);
    }
    // Phase 1: QKV GEMMs (WMMA fp32)
    {
        dim3 grid(ROW_TILES, 3);
        qkv_gemm_kernel<<<grid, 128, 0, stream>>>(X, Wq, Wk, Wv, Q, K, V);
    }
    // Phase 2: edge scores + segment max (8 lanes/edge)
    {
        const long threads = (long)N_EDGES * 8;
        edge_score_kernel<<<(int)((threads + 255) / 256), 256, 0, stream>>>(
            Q, K, src, dst, score, segmax);
    }
    // Phase 3: fused exp + weighted scatter (16 lanes/edge)
    {
        const long threads = (long)N_EDGES * 16;
        edge_aggregate_kernel<<<(int)((threads + 255) / 256), 256, 0, stream>>>(
            V, src, dst, score, segmax, denom, out);
    }
    // Phase 4: normalize
    {
        const int total = N_NODES * D;
        finalize_kernel<<<(total + 255) / 256, 256, 0, stream>>>(out, denom);
    }
}